// RobustRotationEstimatorTorch_36928128811544
// MI455X (gfx1250) — compile-verified
//
#include <hip/hip_runtime.h>

// ---------------------------------------------------------------------------
// Robust rotation estimator (Hough voting) for MI455X / gfx1250.
//
// vote(p,k) = (a_p + (x_p/F)*t_k, b_p + (y_p/F)*t_k, t_k)   [vs=cross(eu,ev) ∝ (x,y,F)]
// bin index = floor(1000*vote + 50.5)  -> 101^3 histogram -> argmax -> mb[idx]
//
// The affine bin transform is folded into V_WMMA_F32_16X16X4_F32:
//   A(16x4) rows = pixels:   [1000*a_p, 1000*(x_p/F), 0, 0]
//   B(4x16)  cols = rots:    [1, t_k, 0, 0]^T           (shared by X and Y WMMA)
//   C       = 50.5 broadcast
//   D[p,k]  = pre-floor bin coordinate (256 votes per WMMA)
// Kernel is L2-atomic bound (13.2M atomics into a 4.12MB L2-resident histogram).
// ---------------------------------------------------------------------------

typedef __attribute__((ext_vector_type(2))) float v2f;
typedef __attribute__((ext_vector_type(8))) float v8f;

#define FOC    500.0f
#define NB     101
#define NB2    (NB * NB)
#define NB3    (NB * NB * NB)      // 1,030,301 bins
#define NROTV  202
#define NTILES 13                  // ceil(202/16)

// -------------------------- init: zero workspace ---------------------------
__global__ void init_kernel(unsigned int* __restrict__ hist,
                            unsigned long long* __restrict__ best)
{
    const int i = blockIdx.x * blockDim.x + threadIdx.x;
    if (i < NB3) hist[i] = 0u;
    if (i == 0)  *best = 0ull;     // key for (count=0); any real bin beats it
}

// ----------------------------- vote kernel ---------------------------------
// 4096 waves, each handles 16 consecutive pixels x 202 rotation samples.
__global__ __launch_bounds__(256) void vote_kernel(const float* __restrict__ flow,
                                                   unsigned int* __restrict__ hist)
{
    const int  lane   = threadIdx.x & 31;
    const int  wave   = (blockIdx.x * blockDim.x + threadIdx.x) >> 5;  // 0..4095
    const int  m      = lane & 15;
    const bool loHalf = lane < 16;

    // Pixel on the 256x256 subsampled grid; 16 consecutive columns per wave.
    const int p  = wave * 16 + m;
    const int pr = p >> 8;
    const int pc = p & 255;

    const float x = (float)(4 * pr - 512) + 0.5f;
    const float y = (float)(4 * pc - 512) + 0.5f;

    const float u = flow[(4 * pr) * 1024 + 4 * pc];
    const float v = flow[1024 * 1024 + (4 * pr) * 1024 + 4 * pc];

    const float den = FOC * FOC * FOC + FOC * x * x + FOC * y * y;
    const float a   =  (FOC * FOC * v - u * x * y + v * x * x) / den;
    const float b   = -(FOC * FOC * u + u * y * y - v * x * y) / den;
    const float wx  = x / FOC;     // vs.x/vs.z (normalization cancels)
    const float wy  = y / FOC;

    const float SCALE = 1000.0f;   // 1/BIN
    const float CBIAS = 50.5f;     // MAXA/BIN + 0.5

    // A fragments (16x4 f32): lanes 0-15 carry K=0,1 coeffs; lanes 16-31 carry K=2,3 = 0.
    v2f aX, aY;
    aX.x = loHalf ? a  * SCALE : 0.0f;
    aX.y = loHalf ? wx * SCALE : 0.0f;
    aY.x = loHalf ? b  * SCALE : 0.0f;
    aY.y = loHalf ? wy * SCALE : 0.0f;

    // t_k = linspace(0,1,202)[k] * (MB[100]-MB[0]) + MB[0]
    const float mb0  = -0.05f;
    const float span = (-0.05f + 0.001f * 100.0f) - mb0;

    for (int kt = 0; kt < NTILES; ++kt) {
        const int   k      = kt * 16 + m;         // rotation sample for N = lane&15
        const bool  kvalid = (k < NROTV);
        const float t      = fmaf((float)k * (1.0f / 201.0f), span, mb0);

        // B fragment (4x16 f32): VGPR0 = row K=0 (ones), VGPR1 = row K=1 (t_k),
        // lanes 16-31 hold rows K=2,3 = 0. Shared by both WMMAs.
        v2f bT;
        bT.x = loHalf ? 1.0f : 0.0f;
        bT.y = (loHalf && kvalid) ? t : 0.0f;

        v8f cX, cY;
#pragma unroll
        for (int j = 0; j < 8; ++j) { cX[j] = CBIAS; cY[j] = CBIAS; }

        // D = A*B + C  ->  1000*(a + w*t) + 50.5 (pre-floor bin coordinates)
        cX = __builtin_amdgcn_wmma_f32_16x16x4_f32(false, aX, false, bT,
                                                   (short)0, cX, false, false);
        cY = __builtin_amdgcn_wmma_f32_16x16x4_f32(false, aY, false, bT,
                                                   (short)0, cY, false, false);

        // z-bin for this lane's rotation sample (same for both lane halves).
        const int iz = (int)floorf(fmaf(SCALE, t, CBIAS));

#pragma unroll
        for (int j = 0; j < 8; ++j) {             // D VGPR j: pixel M = j + 8*(lane>=16)
            const int ix = (int)floorf(cX[j]);
            const int iy = (int)floorf(cY[j]);
            if (kvalid && (unsigned)ix < (unsigned)NB &&
                          (unsigned)iy < (unsigned)NB &&
                          (unsigned)iz < (unsigned)NB) {
                atomicAdd(&hist[ix * NB2 + iy * NB + iz], 1u);
            }
        }
    }
}

// ----------------------------- argmax kernel -------------------------------
// key = (count << 32) | (0xFFFFFFFF - idx): ties resolve to smallest idx,
// matching jnp.argmax's first-occurrence semantics. Deterministic.
__global__ __launch_bounds__(256) void argmax_kernel(const unsigned int* __restrict__ hist,
                                                     unsigned long long* __restrict__ best)
{
    __shared__ unsigned long long sred[256];
    unsigned long long lbest = 0ull;
    for (int i = blockIdx.x * blockDim.x + threadIdx.x; i < NB3;
         i += gridDim.x * blockDim.x) {
        const unsigned long long key =
            ((unsigned long long)hist[i] << 32) |
            (unsigned long long)(0xFFFFFFFFu - (unsigned int)i);
        if (key > lbest) lbest = key;
    }
    sred[threadIdx.x] = lbest;
    __syncthreads();
    for (int s = 128; s > 0; s >>= 1) {
        if ((int)threadIdx.x < s) {
            const unsigned long long o = sred[threadIdx.x + s];
            if (o > sred[threadIdx.x]) sred[threadIdx.x] = o;
        }
        __syncthreads();
    }
    if (threadIdx.x == 0) atomicMax(best, sred[0]);
}

// ----------------------------- finalize ------------------------------------
__global__ void finalize_kernel(const unsigned long long* __restrict__ best,
                                float* __restrict__ out)
{
    if (threadIdx.x == 0 && blockIdx.x == 0) {
        const unsigned int idx = 0xFFFFFFFFu - (unsigned int)(*best & 0xFFFFFFFFull);
        const int i0 = (int)(idx / NB2);
        const int r  = (int)(idx % NB2);
        const int i1 = r / NB;
        const int i2 = r % NB;
        // MB[i] = float32(-0.05 + 0.001*i) computed in double, like numpy.
        out[0] = (float)(-0.05 + 0.001 * (double)i0);
        out[1] = (float)(-0.05 + 0.001 * (double)i1);
        out[2] = (float)(-0.05 + 0.001 * (double)i2);
    }
}

// ----------------------------- launcher ------------------------------------
extern "C" void kernel_launch(void* const* d_in, const int* in_sizes, int n_in,
                              void* d_out, int out_size, void* d_ws, size_t ws_size,
                              hipStream_t stream)
{
    const float* flow = (const float*)d_in[0];
    float*       out  = (float*)d_out;

    unsigned int*       hist = (unsigned int*)d_ws;
    unsigned long long* best =
        (unsigned long long*)((char*)d_ws + (((size_t)NB3 * 4 + 7) & ~(size_t)7));

    init_kernel<<<(NB3 + 255) / 256, 256, 0, stream>>>(hist, best);
    // 512 blocks * 8 waves/block = 4096 waves * 16 pixels = 65536 pixels
    vote_kernel<<<512, 256, 0, stream>>>(flow, hist);
    argmax_kernel<<<1024, 256, 0, stream>>>(hist, best);
    finalize_kernel<<<1, 64, 0, stream>>>(best, out);
}